// SpotlightLoss_67920612819397
// MI455X (gfx1250) — compile-verified
//
#include <hip/hip_runtime.h>
#include <hip/hip_bf16.h>
#include <math.h>

// Problem constants (shape (16,1,16,256,256) fp32)
#define N_TOTAL   16777216              // 16*16*256*256
#define S_SAMPLE  1048576               // elements per sample = 1<<20
#define S_SHIFT   20
#define NB        16
#define NBINS     256
#define BLK_AB    128                   // blocks per sample for minmax/hist
#define NBLK_MAX  4096
#define THREADS   256
#define SIG_K     (-0.95f)

// Workspace layout in 32-bit words
#define WS_LO     0                     // 16 uint order-keys (min)
#define WS_HI     16                    // 16 uint order-keys (max)
#define WS_THR    32                    // 16 float thresholds
#define WS_HIST   48                    // 16*256 uint histogram
#define WS_PART   (48 + NB*NBINS)       // NBLK*5 float block partials
#define WS_WORDS_MAX (WS_PART + NBLK_MAX*5)

// ---- order-preserving float<->uint keys for exact integer atomic min/max ----
__device__ __forceinline__ unsigned fkey(float f) {
  unsigned u = __float_as_uint(f);
  return (u & 0x80000000u) ? ~u : (u | 0x80000000u);
}
__device__ __forceinline__ float funkey(unsigned k) {
  unsigned u = (k & 0x80000000u) ? (k ^ 0x80000000u) : ~k;
  return __uint_as_float(u);
}

// ---- WMMA-based wave(32) sum: D = ones(16x4) x B(4x16) + 0 gives column sums
// B VGPR0 carries lane values as K-rows {0, and 1-or-2}; VGPR1 = 0. Every D
// column n = v[n] + v[n+16] (independent of exact K-row interleave). Then 4
// shfl_xor steps finish 16 -> 1. EXEC must be all ones at call sites.
typedef float v2f __attribute__((ext_vector_type(2)));
typedef float v8f __attribute__((ext_vector_type(8)));

__device__ __forceinline__ float wave_sum_wmma(float v) {
  v2f a; a[0] = 1.0f; a[1] = 1.0f;      // A = ones (16x4)
  v2f b; b[0] = v;    b[1] = 0.0f;      // lane values in first B VGPR
  v8f c = {};
  c = __builtin_amdgcn_wmma_f32_16x16x4_f32(false, a, false, b,
                                            (short)0, c, false, false);
  float s = c[0];                       // s = v[n] + v[n+16], n = lane & 15
  s += __shfl_xor(s, 8, 32);
  s += __shfl_xor(s, 4, 32);
  s += __shfl_xor(s, 2, 32);
  s += __shfl_xor(s, 1, 32);
  return s;                             // full 32-lane sum in every lane
}

// ---------------- kernels ----------------
__global__ void k_init(unsigned* __restrict__ ws, int nwords) {
  int i = blockIdx.x * blockDim.x + threadIdx.x;
  int stride = gridDim.x * blockDim.x;
  for (int j = i; j < nwords; j += stride) {
    unsigned v = 0u;
    if (j < 16) v = 0xFFFFFFFFu;        // lo keys start at +max
    ws[j] = v;                          // hi keys / hist / thr / partials -> 0
  }
}

__global__ void k_minmax(const float* __restrict__ tgt, unsigned* __restrict__ ws) {
  const int b = blockIdx.y;
  const float4* p = (const float4*)(tgt + (size_t)b * S_SAMPLE);
  const int n4 = S_SAMPLE / 4;
  int tid = blockIdx.x * blockDim.x + threadIdx.x;
  int stride = gridDim.x * blockDim.x;
  float lo = INFINITY, hi = -INFINITY;
  for (int i = tid; i < n4; i += stride) {
    float4 v = p[i];
    lo = fminf(lo, fminf(fminf(v.x, v.y), fminf(v.z, v.w)));
    hi = fmaxf(hi, fmaxf(fmaxf(v.x, v.y), fmaxf(v.z, v.w)));
  }
  #pragma unroll
  for (int m = 16; m >= 1; m >>= 1) {
    lo = fminf(lo, __shfl_xor(lo, m, 32));
    hi = fmaxf(hi, __shfl_xor(hi, m, 32));
  }
  __shared__ float slo[8], shi[8];
  int lane = threadIdx.x & 31, wv = threadIdx.x >> 5;
  if (lane == 0) { slo[wv] = lo; shi[wv] = hi; }
  __syncthreads();
  if (threadIdx.x == 0) {
    for (int w = 1; w < 8; w++) { lo = fminf(lo, slo[w]); hi = fmaxf(hi, shi[w]); }
    atomicMin(&ws[WS_LO + b], fkey(lo));
    atomicMax(&ws[WS_HI + b], fkey(hi));
  }
}

__global__ void k_hist(const float* __restrict__ tgt, unsigned* __restrict__ ws) {
  const int b = blockIdx.y;
  __shared__ unsigned lh[NBINS];
  for (int i = threadIdx.x; i < NBINS; i += blockDim.x) lh[i] = 0u;
  __syncthreads();
  const float lo = funkey(ws[WS_LO + b]);
  const float hi = funkey(ws[WS_HI + b]);
  const float span = hi - lo;
  const float ss = (span > 0.0f) ? span : 1.0f;
  const float4* p = (const float4*)(tgt + (size_t)b * S_SAMPLE);
  const int n4 = S_SAMPLE / 4;
  int tid = blockIdx.x * blockDim.x + threadIdx.x;
  int stride = gridDim.x * blockDim.x;
  for (int i = tid; i < n4; i += stride) {
    float4 v = p[i];
    float vv[4] = {v.x, v.y, v.z, v.w};
    #pragma unroll
    for (int j = 0; j < 4; j++) {
      int k = (int)floorf(((vv[j] - lo) / ss) * 256.0f);   // matches reference math
      k = k < 0 ? 0 : (k > 255 ? 255 : k);
      atomicAdd(&lh[k], 1u);
    }
  }
  __syncthreads();
  for (int i = threadIdx.x; i < NBINS; i += blockDim.x)
    if (lh[i]) atomicAdd(&ws[WS_HIST + b * NBINS + i], lh[i]);
}

__global__ void k_otsu(unsigned* __restrict__ ws) {
  const int b = blockIdx.x;
  __shared__ float cum_s[NBINS], cum_m[NBINS];
  if (threadIdx.x != 0) return;         // tiny: serial, matches jnp.cumsum order
  const float lo = funkey(ws[WS_LO + b]);
  const float hi = funkey(ws[WS_HI + b]);
  const float span = hi - lo;
  const unsigned* h = &ws[WS_HIST + b * NBINS];
  const float inv = 1.0f / 256.0f;
  float rs = 0.0f, rm = 0.0f;
  for (int i = 0; i < NBINS; i++) {
    float hf = (float)h[i];
    float c = lo + span * ((i + 0.5f) * inv);
    rs += hf;
    rm += hf * c;
    cum_s[i] = rs; cum_m[i] = rm;
  }
  const float total = rs;               // integer counts: exact in fp32
  const float gm = rm;
  float best = -INFINITY; int bi = 0;
  for (int i = 0; i < NBINS; i++) {
    float w0 = cum_s[i];
    float w1 = total - w0;
    float num = cum_m[i] * total - gm * w0;
    float icv = (num * num) / (w0 * w1 + 1e-10f);
    if (icv > best) { best = icv; bi = i; }   // first-max, like jnp.argmax
  }
  float thr = (span > 0.0f) ? (lo + span * ((bi + 0.5f) * inv)) : lo;
  ((float*)ws)[WS_THR + b] = thr;
}

__global__ void k_main(const float* __restrict__ pred, const float* __restrict__ tgt,
                       const unsigned* __restrict__ ws, float* __restrict__ part) {
  const float* thr_arr = (const float*)&ws[WS_THR];
  const float4* p4 = (const float4*)pred;
  const float4* t4 = (const float4*)tgt;
  const int n4 = N_TOTAL / 4;
  int tid = blockIdx.x * blockDim.x + threadIdx.x;
  int stride = gridDim.x * blockDim.x;
  float msq = 0.f, mcnt = 0.f, ssum = 0.f, inter = 0.f, tsq = 0.f;
  for (int i = tid; i < n4; i += stride) {
    const int samp = i >> (S_SHIFT - 2);      // float4 never crosses a sample
    const float thr = thr_arr[samp];
    float4 pv = p4[i];
    float4 tv = t4[i];
    float pa[4] = {pv.x, pv.y, pv.z, pv.w};
    float ta[4] = {tv.x, tv.y, tv.z, tv.w};
    #pragma unroll
    for (int j = 0; j < 4; j++) {
      float t = ta[j], x = pa[j];
      float m = (t >= thr) ? 1.0f : 0.0f;
      float d = x - t;
      float sq = d * d;
      tsq  += sq;
      msq  += sq * m;
      mcnt += m;
      float ax = fabsf(x);
      float raw = (x - SIG_K * x) / ((SIG_K - 2.0f * SIG_K * ax) + 1.0f);
      float sp = fminf(fmaxf(raw, 0.0f), 1.0f);
      ssum  += sp;
      inter += sp * m;
    }
  }
  float q[5] = {msq, mcnt, ssum, inter, tsq};
  __shared__ float sblk[8][5];
  int lane = threadIdx.x & 31, wv = threadIdx.x >> 5;
  #pragma unroll
  for (int j = 0; j < 5; j++) {
    float s = wave_sum_wmma(q[j]);
    if (lane == 0) sblk[wv][j] = s;
  }
  __syncthreads();
  if (threadIdx.x == 0) {
    #pragma unroll
    for (int j = 0; j < 5; j++) {
      float s = 0.f;
      for (int w = 0; w < 8; w++) s += sblk[w][j];
      part[blockIdx.x * 5 + j] = s;           // no float atomics: deterministic
    }
  }
}

__global__ void k_final(const float* __restrict__ part, int nblk, float* __restrict__ out) {
  float q[5] = {0.f, 0.f, 0.f, 0.f, 0.f};
  for (int b = threadIdx.x; b < nblk; b += blockDim.x) {
    #pragma unroll
    for (int j = 0; j < 5; j++) q[j] += part[b * 5 + j];
  }
  __shared__ float sblk[8][5];
  int lane = threadIdx.x & 31, wv = threadIdx.x >> 5;
  #pragma unroll
  for (int j = 0; j < 5; j++) {
    float s = wave_sum_wmma(q[j]);
    if (lane == 0) sblk[wv][j] = s;
  }
  __syncthreads();
  if (threadIdx.x == 0) {
    float tot[5];
    #pragma unroll
    for (int j = 0; j < 5; j++) {
      float s = 0.f;
      for (int w = 0; w < 8; w++) s += sblk[w][j];
      tot[j] = s;
    }
    float msq = tot[0], mcnt = tot[1], ssum = tot[2], inter = tot[3], tsq = tot[4];
    float masked_mse = (mcnt > 0.0f) ? (msq / mcnt) : (tsq / (float)N_TOTAL);
    float dice = 1.0f - (2.0f * inter) / (ssum + mcnt + 1e-6f);
    out[0] = 0.5f * masked_mse + 0.5f * dice;
  }
}

extern "C" void kernel_launch(void* const* d_in, const int* in_sizes, int n_in,
                              void* d_out, int out_size, void* d_ws, size_t ws_size,
                              hipStream_t stream) {
  (void)in_sizes; (void)n_in; (void)out_size;
  const float* pred = (const float*)d_in[0];
  const float* tgt  = (const float*)d_in[1];
  float* out = (float*)d_out;
  unsigned* ws = (unsigned*)d_ws;
  float* part = (float*)(ws + WS_PART);

  // Size the block-partial region to the workspace we actually have.
  long wswords = (long)(ws_size / 4);
  long avail = wswords - WS_PART;
  int nblk = (int)(avail / 5);
  if (nblk > NBLK_MAX) nblk = NBLK_MAX;
  if (nblk < 1) nblk = 1;
  int nwords = WS_PART + nblk * 5;
  if (nwords > wswords) nwords = (int)wswords;

  k_init<<<32, THREADS, 0, stream>>>(ws, nwords);
  dim3 gAB(BLK_AB, NB);
  k_minmax<<<gAB, THREADS, 0, stream>>>(tgt, ws);
  k_hist<<<gAB, THREADS, 0, stream>>>(tgt, ws);
  k_otsu<<<NB, 32, 0, stream>>>(ws);
  k_main<<<nblk, THREADS, 0, stream>>>(pred, tgt, ws, part);
  k_final<<<1, THREADS, 0, stream>>>(part, nblk, out);
}